// DummyTeGG_37632503447554
// MI455X (gfx1250) — compile-verified
//
#include <hip/hip_runtime.h>
#include <hip/hip_bf16.h>

// Grouped MLP: out = (x @ W1^T + b1) @ W2^T + b2, 8 groups.
// Full-precision path via V_WMMA_F32_16X16X4_F32 (exact fp32 math).
// WG tile 128x128, 8 wave32s each computing a 64x32 C tile (4x2 WMMA accums).

typedef __attribute__((ext_vector_type(2))) float v2f;
typedef __attribute__((ext_vector_type(8))) float v8f;

#define WG_M 128         // workgroup tile rows (M)
#define WG_N 128         // workgroup tile cols (N)
#define KT   16          // K slab per LDS stage
#define LDS_STRIDE 20    // KT + 4 pad: 16B-aligned float4 stores, conflict-free b64 reads

// C[M,N] = A[M,K] * B[N,K]^T + bias[N], one group per blockIdx.z.
// A, B are both K-contiguous (torch Linear convention), so no transpose needed.
__global__ __launch_bounds__(256, 2)
void grouped_gemm_bias_f32_wmma(const float* __restrict__ Aall,
                                const float* __restrict__ Ball,
                                const float* __restrict__ biasAll,
                                float* __restrict__ Call,
                                int M, int N, int K) {
    const float* A    = Aall    + (size_t)blockIdx.z * (size_t)M * (size_t)K;
    const float* B    = Ball    + (size_t)blockIdx.z * (size_t)N * (size_t)K;
    const float* bias = biasAll + (size_t)blockIdx.z * (size_t)N;
    float*       C    = Call    + (size_t)blockIdx.z * (size_t)M * (size_t)N;

    const int m0 = blockIdx.y * WG_M;
    const int n0 = blockIdx.x * WG_N;

    __shared__ __align__(16) float sA[WG_M * LDS_STRIDE];   // 10240 B
    __shared__ __align__(16) float sB[WG_N * LDS_STRIDE];   // 10240 B

    const int t    = threadIdx.x;
    const int lane = t & 31;          // wave32
    const int wid  = t >> 5;          // 0..7
    const int wm   = wid & 1;         // wave row block: 2 x 64 rows
    const int wn   = wid >> 1;        // wave col block: 4 x 32 cols
    const int lr   = lane & 15;       // M/N index within a 16-wide fragment
    const int lh   = lane >> 4;       // K half: lanes 16-31 hold K+2,K+3

    // --- global staging assignments (256 threads) ---
    // A tile 128x16 floats: thread -> row t>>1, 8 floats at (t&1)*8 (two float4)
    // B tile 128x16 floats: same pattern
    const int sRow = t >> 1;
    const int sCol = (t & 1) * 8;

    const float* aPtr = A + (size_t)(m0 + sRow) * (size_t)K + sCol;
    const float* bPtr = B + (size_t)(n0 + sRow) * (size_t)K + sCol;

    // prefetch slab 0 into registers
    float4 ra0 = *(const float4*)(aPtr);
    float4 ra1 = *(const float4*)(aPtr + 4);
    float4 rb0 = *(const float4*)(bPtr);
    float4 rb1 = *(const float4*)(bPtr + 4);

    v8f acc[4][2];
    #pragma unroll
    for (int i = 0; i < 4; ++i)
        #pragma unroll
        for (int j = 0; j < 2; ++j)
            acc[i][j] = (v8f)(0.0f);

    const int ksteps = K / KT;
    for (int kt = 0; kt < ksteps; ++kt) {
        // commit staged registers to LDS
        *(float4*)&sA[sRow * LDS_STRIDE + sCol]     = ra0;
        *(float4*)&sA[sRow * LDS_STRIDE + sCol + 4] = ra1;
        *(float4*)&sB[sRow * LDS_STRIDE + sCol]     = rb0;
        *(float4*)&sB[sRow * LDS_STRIDE + sCol + 4] = rb1;
        __syncthreads();

        // prefetch next slab while we compute on this one
        if (kt + 1 < ksteps) {
            const float* ap = aPtr + (size_t)(kt + 1) * KT;
            const float* bp = bPtr + (size_t)(kt + 1) * KT;
            ra0 = *(const float4*)(ap);
            ra1 = *(const float4*)(ap + 4);
            rb0 = *(const float4*)(bp);
            rb1 = *(const float4*)(bp + 4);
        }

        // 4 WMMA K-steps of 4 over this slab: 32 v_wmma per wave per slab
        #pragma unroll
        for (int kk = 0; kk < 4; ++kk) {
            const int kOff = kk * 4 + lh * 2;   // lanes 0-15: K+{0,1}; lanes 16-31: K+{2,3}
            v2f aF[4], bF[2];
            #pragma unroll
            for (int mi = 0; mi < 4; ++mi)
                aF[mi] = *(const v2f*)&sA[(wm * 64 + mi * 16 + lr) * LDS_STRIDE + kOff];
            #pragma unroll
            for (int ni = 0; ni < 2; ++ni)
                bF[ni] = *(const v2f*)&sB[(wn * 32 + ni * 16 + lr) * LDS_STRIDE + kOff];
            #pragma unroll
            for (int mi = 0; mi < 4; ++mi)
                #pragma unroll
                for (int ni = 0; ni < 2; ++ni)
                    acc[mi][ni] = __builtin_amdgcn_wmma_f32_16x16x4_f32(
                        /*neg_a=*/false, aF[mi],
                        /*neg_b=*/false, bF[ni],
                        /*c_mod=*/(short)0, acc[mi][ni],
                        /*reuse_a=*/false, /*reuse_b=*/false);
        }
        __syncthreads();
    }

    // epilogue: bias + store.  C layout: VGPR r -> row r + 8*lh, col = lane%16.
    #pragma unroll
    for (int ni = 0; ni < 2; ++ni) {
        const int col = n0 + wn * 32 + ni * 16 + lr;
        const float bv = bias[col];
        #pragma unroll
        for (int mi = 0; mi < 4; ++mi) {
            const int rowBase = m0 + wm * 64 + mi * 16 + lh * 8;
            #pragma unroll
            for (int r = 0; r < 8; ++r)
                C[(size_t)(rowBase + r) * (size_t)N + col] = acc[mi][ni][r] + bv;
        }
    }
}

extern "C" void kernel_launch(void* const* d_in, const int* in_sizes, int n_in,
                              void* d_out, int out_size, void* d_ws, size_t ws_size,
                              hipStream_t stream) {
    (void)in_sizes; (void)n_in; (void)out_size; (void)ws_size;
    const float* x  = (const float*)d_in[0];   // [8, 2048, 1024]
    const float* W1 = (const float*)d_in[1];   // [8, 4096, 1024]
    const float* b1 = (const float*)d_in[2];   // [8, 4096]
    const float* W2 = (const float*)d_in[3];   // [8, 1024, 4096]
    const float* b2 = (const float*)d_in[4];   // [8, 1024]
    float* out = (float*)d_out;                // [8, 2048, 1024]
    float* h1  = (float*)d_ws;                 // [8, 2048, 4096] fp32 = 256 MB scratch

    const int G = 8, Mm = 2048, H = 1024, I = 4096;

    // GEMM1: h1 = x @ W1^T + b1   (M=2048, N=4096, K=1024)
    dim3 g1(I / WG_N, Mm / WG_M, G);
    grouped_gemm_bias_f32_wmma<<<g1, 256, 0, stream>>>(x, W1, b1, h1, Mm, I, H);

    // GEMM2: out = h1 @ W2^T + b2 (M=2048, N=1024, K=4096)
    dim3 g2(H / WG_N, Mm / WG_M, G);
    grouped_gemm_bias_f32_wmma<<<g2, 256, 0, stream>>>(h1, W2, b2, out, Mm, H, I);
}